// PseudoGenerator_13597866459481
// MI455X (gfx1250) — compile-verified
//
#include <hip/hip_runtime.h>
#include <math.h>

#define B_    16
#define C_    512
#define N_    4096
#define TOPK_ 12

typedef __attribute__((ext_vector_type(2))) float v2f;
typedef __attribute__((ext_vector_type(8))) float v8f;

// ---------------- workspace layout (float offsets) ----------------
#define WS_RES_FG   0            // B*N
#define WS_RES_BG   65536
#define WS_DIN_FG   131072
#define WS_DIN_BG   196608
#define WS_MASK_FG  262144       // B*N (0/1 floats)
#define WS_MASK_BG  327680
#define WS_RES_FP   393216       // B*C protos
#define WS_DIN_FP   401408
#define WS_RES_BP   409600
#define WS_DIN_BP   417792
#define WS_PNORM    425984       // B*4 proto norms {res_fp,res_bp,din_fp,din_bp}
#define WS_TOPK     426048       // int[4*B*12]
#define WS_CNT      426816       // int[2*B] : cnt_fg, cnt_bg

// ---------------- output layout (float offsets) ----------------
#define O_RESQ   0               // (B,2,N)
#define O_DINQ   131072
#define O_FUSEFP 262144          // (B,C)
#define O_FUSEBP 270336
#define O_MEANFP 278528
#define O_LOGVFP 286720
#define O_MEANBP 294912
#define O_LOGVBP 303104

__device__ __forceinline__ void fma44(float4& a, const float4& x, const float4& y) {
  a.x += x.x * y.x; a.y += x.y * y.y; a.z += x.z * y.z; a.w += x.w * y.w;
}
__device__ __forceinline__ void fma4s(float4& a, const float4& x, float s) {
  a.x += x.x * s; a.y += x.y * s; a.z += x.z * s; a.w += x.w * s;
}

// ================= K0: proto L2 norms per batch =================
__global__ void k0_proto_norms(const float* __restrict__ rfp, const float* __restrict__ rbp,
                               const float* __restrict__ dfp, const float* __restrict__ dbp,
                               float* __restrict__ ws) {
  int b = blockIdx.x, t = threadIdx.x;
  __shared__ float red[4][256];
  float a0 = 0, a1 = 0, a2 = 0, a3 = 0;
  for (int c = t; c < C_; c += 256) {
    float x;
    x = rfp[b * C_ + c]; a0 += x * x;
    x = rbp[b * C_ + c]; a1 += x * x;
    x = dfp[b * C_ + c]; a2 += x * x;
    x = dbp[b * C_ + c]; a3 += x * x;
  }
  red[0][t] = a0; red[1][t] = a1; red[2][t] = a2; red[3][t] = a3;
  __syncthreads();
  for (int s = 128; s > 0; s >>= 1) {
    if (t < s) {
      red[0][t] += red[0][t + s]; red[1][t] += red[1][t + s];
      red[2][t] += red[2][t + s]; red[3][t] += red[3][t + s];
    }
    __syncthreads();
  }
  if (t == 0) {
    float* pn = ws + WS_PNORM + b * 4;
    pn[0] = sqrtf(red[0][0]); pn[1] = sqrtf(red[1][0]);
    pn[2] = sqrtf(red[2][0]); pn[3] = sqrtf(red[3][0]);
  }
}

// ===== K1: fused similarity + softmax + masks (pass 1 over features) =====
__global__ void k1_similarity(const float* __restrict__ resq, const float* __restrict__ dinq,
                              const float* __restrict__ rfp, const float* __restrict__ rbp,
                              const float* __restrict__ dfp, const float* __restrict__ dbp,
                              const float* __restrict__ fg_thres_p,
                              const float* __restrict__ bg_thres_p,
                              float* __restrict__ out, float* __restrict__ ws) {
  int b = blockIdx.y, t = threadIdx.x;
  int n0 = blockIdx.x * 1024 + t * 4;            // 4 pixels per thread (float4)
  __shared__ float sp[4][C_];                    // protos: 8 KB
  for (int c = t; c < C_; c += 256) {
    sp[0][c] = rfp[b * C_ + c]; sp[1][c] = rbp[b * C_ + c];
    sp[2][c] = dfp[b * C_ + c]; sp[3][c] = dbp[b * C_ + c];
  }
  __syncthreads();
  const float4* pr = (const float4*)(resq + (size_t)b * C_ * N_ + n0);
  const float4* pd = (const float4*)(dinq + (size_t)b * C_ * N_ + n0);
  float4 drf = {0,0,0,0}, drb = {0,0,0,0}, drr = {0,0,0,0};
  float4 ddf = {0,0,0,0}, ddb = {0,0,0,0}, ddd = {0,0,0,0};
  for (int c = 0; c < C_; ++c) {
    float4 r = pr[(size_t)c * (N_ / 4)];
    float4 d = pd[(size_t)c * (N_ / 4)];
    fma4s(drf, r, sp[0][c]); fma4s(drb, r, sp[1][c]); fma44(drr, r, r);
    fma4s(ddf, d, sp[2][c]); fma4s(ddb, d, sp[3][c]); fma44(ddd, d, d);
  }
  const float* pn = ws + WS_PNORM + b * 4;
  float n_rf = pn[0], n_rb = pn[1], n_df = pn[2], n_db = pn[3];
  float fg_t = *fg_thres_p, bg_t = *bg_thres_p;
  float RF[4] = {drf.x,drf.y,drf.z,drf.w}, RB[4] = {drb.x,drb.y,drb.z,drb.w};
  float RR[4] = {drr.x,drr.y,drr.z,drr.w};
  float DF[4] = {ddf.x,ddf.y,ddf.z,ddf.w}, DB[4] = {ddb.x,ddb.y,ddb.z,ddb.w};
  float DD[4] = {ddd.x,ddd.y,ddd.z,ddd.w};
  for (int k = 0; k < 4; ++k) {
    int n = n0 + k;
    float nr = sqrtf(RR[k]), nd = sqrtf(DD[k]);
    float s_rf = 10.0f * RF[k] / fmaxf(nr * n_rf, 1e-8f);
    float s_rb = 10.0f * RB[k] / fmaxf(nr * n_rb, 1e-8f);
    float s_df = 10.0f * DF[k] / fmaxf(nd * n_df, 1e-8f);
    float s_db = 10.0f * DB[k] / fmaxf(nd * n_db, 1e-8f);
    out[O_RESQ + b * 2 * N_ + n]      = s_rb;   // channel 0 = bg
    out[O_RESQ + b * 2 * N_ + N_ + n] = s_rf;   // channel 1 = fg
    out[O_DINQ + b * 2 * N_ + n]      = s_db;
    out[O_DINQ + b * 2 * N_ + N_ + n] = s_df;
    float pf_r = 1.0f / (1.0f + expf(s_rb - s_rf));
    float pb_r = 1.0f / (1.0f + expf(s_rf - s_rb));
    float pf_d = 1.0f / (1.0f + expf(s_db - s_df));
    float pb_d = 1.0f / (1.0f + expf(s_df - s_db));
    ws[WS_RES_FG + b * N_ + n] = pf_r;
    ws[WS_RES_BG + b * N_ + n] = pb_r;
    ws[WS_DIN_FG + b * N_ + n] = pf_d;
    ws[WS_DIN_BG + b * N_ + n] = pb_d;
    ws[WS_MASK_FG + b * N_ + n] = (pf_r > fg_t && pf_d > fg_t) ? 1.0f : 0.0f;
    ws[WS_MASK_BG + b * N_ + n] = (pb_r > bg_t && pb_d > bg_t) ? 1.0f : 0.0f;
  }
}

// ================= K2: top-12 indices + mask counts =================
__global__ void k2_topk(float* __restrict__ ws) {
  int type = blockIdx.x;   // 0 res_fg, 1 din_fg, 2 res_bg, 3 din_bg
  int b = blockIdx.y, t = threadIdx.x;
  __shared__ float vals[N_];
  __shared__ float rv[256];
  __shared__ int   ri[256];
  int soff = (type == 0) ? WS_RES_FG : (type == 1) ? WS_DIN_FG
           : (type == 2) ? WS_RES_BG : WS_DIN_BG;
  const float* src = ws + soff + b * N_;
  for (int i = t; i < N_; i += 256) vals[i] = src[i];
  // mask count
  const float* msk = ws + ((type < 2) ? WS_MASK_FG : WS_MASK_BG) + b * N_;
  float cs = 0;
  for (int i = t; i < N_; i += 256) cs += msk[i];
  rv[t] = cs; __syncthreads();
  for (int s = 128; s > 0; s >>= 1) { if (t < s) rv[t] += rv[t + s]; __syncthreads(); }
  if (t == 0) {
    int* cnt = (int*)(ws + WS_CNT);
    if (type == 0) cnt[b]      = (int)(rv[0] + 0.5f);
    if (type == 2) cnt[B_ + b] = (int)(rv[0] + 0.5f);
  }
  __syncthreads();
  int* tk = (int*)(ws + WS_TOPK) + (type * B_ + b) * TOPK_;
  for (int it = 0; it < TOPK_; ++it) {
    float bv = -3.4e38f; int bi = N_;
    for (int i = t; i < N_; i += 256) {
      float v = vals[i];
      if (v > bv || (v == bv && i < bi)) { bv = v; bi = i; }
    }
    rv[t] = bv; ri[t] = bi; __syncthreads();
    for (int s = 128; s > 0; s >>= 1) {
      if (t < s) {
        float v2 = rv[t + s]; int i2 = ri[t + s];
        if (v2 > rv[t] || (v2 == rv[t] && i2 < ri[t])) { rv[t] = v2; ri[t] = i2; }
      }
      __syncthreads();
    }
    if (t == 0) { tk[it] = ri[0]; vals[ri[0]] = -3.4e38f; }
    __syncthreads();
  }
}

// ===== K3: masked means / top-k means -> protos (pass 2 over features) =====
__global__ void k3_protos(const float* __restrict__ resq, const float* __restrict__ dinq,
                          float* __restrict__ ws) {
  int c = blockIdx.x, b = blockIdx.y, t = threadIdx.x;
  __shared__ float4 red[256];
  const float* rp = resq + ((size_t)(b * C_ + c)) * N_;
  const float* dp = dinq + ((size_t)(b * C_ + c)) * N_;
  const float* mf = ws + WS_MASK_FG + b * N_;
  const float* mb = ws + WS_MASK_BG + b * N_;
  float sfr = 0, sbr = 0, sfd = 0, sbd = 0;
  for (int i = t * 4; i < N_; i += 1024) {
    float4 r = *(const float4*)(rp + i);
    float4 d = *(const float4*)(dp + i);
    float4 f = *(const float4*)(mf + i);
    float4 g = *(const float4*)(mb + i);
    sfr += r.x*f.x + r.y*f.y + r.z*f.z + r.w*f.w;
    sbr += r.x*g.x + r.y*g.y + r.z*g.z + r.w*g.w;
    sfd += d.x*f.x + d.y*f.y + d.z*f.z + d.w*f.w;
    sbd += d.x*g.x + d.y*g.y + d.z*g.z + d.w*g.w;
  }
  red[t] = make_float4(sfr, sbr, sfd, sbd);
  __syncthreads();
  for (int s = 128; s > 0; s >>= 1) {
    if (t < s) {
      red[t].x += red[t+s].x; red[t].y += red[t+s].y;
      red[t].z += red[t+s].z; red[t].w += red[t+s].w;
    }
    __syncthreads();
  }
  if (t == 0) {
    int cf = ((const int*)(ws + WS_CNT))[b];
    int cb = ((const int*)(ws + WS_CNT))[B_ + b];
    const int* tk = (const int*)(ws + WS_TOPK);
    float t0 = 0, t1 = 0, t2 = 0, t3 = 0;
    for (int j = 0; j < TOPK_; ++j) {
      t0 += rp[tk[(0 * B_ + b) * TOPK_ + j]];
      t1 += dp[tk[(1 * B_ + b) * TOPK_ + j]];
      t2 += rp[tk[(2 * B_ + b) * TOPK_ + j]];
      t3 += dp[tk[(3 * B_ + b) * TOPK_ + j]];
    }
    const float inv12 = 1.0f / 12.0f;
    ws[WS_RES_FP + b * C_ + c] = (cf > 0) ? red[0].x / (float)cf : t0 * inv12;
    ws[WS_DIN_FP + b * C_ + c] = (cf > 0) ? red[0].z / (float)cf : t1 * inv12;
    ws[WS_RES_BP + b * C_ + c] = (cb > 0) ? red[0].y / (float)cb : t2 * inv12;
    ws[WS_DIN_BP + b * C_ + c] = (cb > 0) ? red[0].w / (float)cb : t3 * inv12;
  }
}

// ===== K4: mean/logvar GEMMs via V_WMMA_F32_16X16X4_F32 (one wave / tile) =====
// Uniform loop counter (no divergent trip count -> no exec-mask churn around WMMA);
// lane-dependence only in addressing. Unrolled x4 so loads clause up.
__global__ void k4_gemm(const float* __restrict__ Wm_fp, const float* __restrict__ Wl_fp,
                        const float* __restrict__ Wm_bp, const float* __restrict__ Wl_bp,
                        const float* __restrict__ a_fp_p, const float* __restrict__ b_fp_p,
                        const float* __restrict__ a_bp_p, const float* __restrict__ b_bp_p,
                        float* __restrict__ out, const float* __restrict__ ws) {
  int jt = blockIdx.x;        // N tile: 0..31
  int g  = blockIdx.y;        // 0=mean_fp 1=logvar_fp 2=mean_bp 3=logvar_bp
  int lane = threadIdx.x;     // 0..31 (one wave32)
  int half = lane >> 4, l = lane & 15;
  const float* Wmat = (g == 0) ? Wm_fp : (g == 1) ? Wl_fp : (g == 2) ? Wm_bp : Wl_bp;
  const float* rp  = ws + ((g < 2) ? WS_RES_FP : WS_RES_BP);
  const float* dpr = ws + ((g < 2) ? WS_DIN_FP : WS_DIN_BP);
  float alpha = (g < 2) ? *a_fp_p : *a_bp_p;
  float beta  = (g < 2) ? *b_fp_p : *b_bp_p;
  int n0 = jt * 16;
  // Per-lane base pointers already offset by the lane's K phase (2*half):
  const float* wrow   = Wmat + (size_t)(n0 + l) * C_ + 2 * half; // B[k,n]=W[n0+n,k]
  const float* arow_r = rp  + l * C_ + 2 * half;                 // A[m,k]=f[m,k]
  const float* arow_d = dpr + l * C_ + 2 * half;
  v8f acc = {};
  #pragma unroll 4
  for (int kk = 0; kk < C_ / 4; ++kk) {          // uniform trip count: 128
    int c0 = 4 * kk;
    v2f a, w;
    a.x = alpha * arow_r[c0]     + beta * arow_d[c0];
    a.y = alpha * arow_r[c0 + 1] + beta * arow_d[c0 + 1];
    w.x = wrow[c0]; w.y = wrow[c0 + 1];
    acc = __builtin_amdgcn_wmma_f32_16x16x4_f32(false, a, false, w,
                                                (short)0, acc, false, false);
  }
  int ooff = (g == 0) ? O_MEANFP : (g == 1) ? O_LOGVFP : (g == 2) ? O_MEANBP : O_LOGVBP;
  float* o = out + ooff;
  #pragma unroll
  for (int r = 0; r < 8; ++r) {
    int row = r + 8 * half;                  // D: vgpr r -> M=r (lanes<16) / M=r+8
    o[row * C_ + n0 + l] = acc[r];
  }
}

// ================= K5: threefry MC variance + fused output =================
__device__ __forceinline__ float u2f_(unsigned u) { union { unsigned u; float f; } x; x.u = u; return x.f; }

#define TFRND(r) { x0 += x1; x1 = (x1 << (r)) | (x1 >> (32 - (r))); x1 ^= x0; }
__device__ __forceinline__ void tf2x32(unsigned k0, unsigned k1, unsigned x0, unsigned x1,
                                       unsigned& o0, unsigned& o1) {
  unsigned k2 = k0 ^ k1 ^ 0x1BD11BDAu;
  x0 += k0; x1 += k1;
  TFRND(13) TFRND(15) TFRND(26) TFRND(6)  x0 += k1; x1 += k2 + 1u;
  TFRND(17) TFRND(29) TFRND(16) TFRND(24) x0 += k2; x1 += k0 + 2u;
  TFRND(13) TFRND(15) TFRND(26) TFRND(6)  x0 += k0; x1 += k1 + 3u;
  TFRND(17) TFRND(29) TFRND(16) TFRND(24) x0 += k1; x1 += k2 + 4u;
  TFRND(13) TFRND(15) TFRND(26) TFRND(6)  x0 += k2; x1 += k0 + 5u;
  o0 = x0; o1 = x1;
}

__device__ __forceinline__ float erfinvf_(float x) {
  float w = -logf((1.0f - x) * (1.0f + x));
  float p;
  if (w < 5.0f) {
    w = w - 2.5f;
    p = 2.81022636e-08f;
    p = 3.43273939e-07f + p * w;  p = -3.5233877e-06f  + p * w;
    p = -4.39150654e-06f + p * w; p = 0.00021858087f   + p * w;
    p = -0.00125372503f + p * w;  p = -0.00417768164f  + p * w;
    p = 0.246640727f + p * w;     p = 1.50140941f      + p * w;
  } else {
    w = sqrtf(w) - 3.0f;
    p = -0.000200214257f;
    p = 0.000100950558f + p * w;  p = 0.00134934322f   + p * w;
    p = -0.00367342844f + p * w;  p = 0.00573950773f   + p * w;
    p = -0.0076224613f + p * w;   p = 0.00943887047f   + p * w;
    p = 1.00167406f + p * w;      p = 2.83297682f      + p * w;
  }
  return p * x;
}

__device__ __forceinline__ float bits_to_normal(unsigned bits) {
  // jax.random.uniform path: bits>>9 | 1.0f -> [1,2) -> [0,1) -> [lo,1) with lo=nextafter(-1,0)
  float u01 = u2f_((bits >> 9) | 0x3f800000u) - 1.0f;
  const float lo = -0.99999994f;
  float u = fmaxf(lo, u01 * (1.0f - lo) + lo);
  return 1.41421356237f * erfinvf_(u);
}

__global__ void k5_gaussian(const float* __restrict__ a_fp_p, const float* __restrict__ b_fp_p,
                            const float* __restrict__ a_bp_p, const float* __restrict__ b_bp_p,
                            float* __restrict__ out, const float* __restrict__ ws) {
  int which = blockIdx.x;   // 0=fp (fold 0), 1=bp (fold 1)
  int b = blockIdx.y;
  int j = threadIdx.x;      // 0..511
  __shared__ float red[512];
  const float* rp = ws + ((which == 0) ? WS_RES_FP : WS_RES_BP);
  const float* dp = ws + ((which == 0) ? WS_DIN_FP : WS_DIN_BP);
  float alpha = (which == 0) ? *a_fp_p : *a_bp_p;
  float beta  = (which == 0) ? *b_fp_p : *b_bp_p;
  int mean_off  = (which == 0) ? O_MEANFP : O_MEANBP;
  int lv_off    = (which == 0) ? O_LOGVFP : O_LOGVBP;
  int fused_off = (which == 0) ? O_FUSEFP : O_FUSEBP;
  float mean = out[mean_off + b * C_ + j];
  float lv   = out[lv_off   + b * C_ + j];
  float stdv = expf(0.5f * lv);
  unsigned fk0, fk1;
  tf2x32(0u, 42u, 0u, (unsigned)which, fk0, fk1);   // fold_in(key(42), which)
  unsigned q = (unsigned)(b * C_ + j);
  float sum = 0.0f, sumsq = 0.0f;
  for (int k = 0; k < 25; ++k) {                    // p and p+204800 share (b,j): k and k+25
    unsigned p = (unsigned)k * 8192u + q;
    unsigned r0, r1;
    tf2x32(fk0, fk1, p, p + 204800u, r0, r1);
    float s0 = mean + bits_to_normal(r0) * stdv;
    float s1 = mean + bits_to_normal(r1) * stdv;
    sum += s0 + s1; sumsq += s0 * s0 + s1 * s1;
  }
  float var = (sumsq - sum * sum * (1.0f / 50.0f)) * (1.0f / 49.0f);  // ddof=1
  red[j] = var * var; __syncthreads();
  for (int s = 256; s > 0; s >>= 1) { if (j < s) red[j] += red[j + s]; __syncthreads(); }
  float unc = 10.0f * var / sqrtf(red[0]);
  float rpv = rp[b * C_ + j], dpv = dp[b * C_ + j];
  float fused = alpha * rpv + beta * dpv;
  out[fused_off + b * C_ + j] = (1.0f - unc) * fused + rpv + dpv;
}

// ============================ launcher ============================
extern "C" void kernel_launch(void* const* d_in, const int* in_sizes, int n_in,
                              void* d_out, int out_size, void* d_ws, size_t ws_size,
                              hipStream_t stream) {
  const float* res_supp_fp = (const float*)d_in[0];
  const float* res_supp_bp = (const float*)d_in[1];
  const float* res_query   = (const float*)d_in[2];
  const float* din_supp_fp = (const float*)d_in[3];
  const float* din_supp_bp = (const float*)d_in[4];
  const float* din_query   = (const float*)d_in[5];
  const float* fg_thres    = (const float*)d_in[6];
  const float* bg_thres    = (const float*)d_in[7];
  const float* alpha_fp    = (const float*)d_in[8];
  const float* beta_fp     = (const float*)d_in[9];
  const float* W_mean_fp   = (const float*)d_in[10];
  const float* W_logvar_fp = (const float*)d_in[11];
  const float* alpha_bp    = (const float*)d_in[12];
  const float* beta_bp     = (const float*)d_in[13];
  const float* W_mean_bp   = (const float*)d_in[14];
  const float* W_logvar_bp = (const float*)d_in[15];
  float* out = (float*)d_out;
  float* ws  = (float*)d_ws;

  k0_proto_norms<<<B_, 256, 0, stream>>>(res_supp_fp, res_supp_bp, din_supp_fp, din_supp_bp, ws);
  k1_similarity<<<dim3(4, B_), 256, 0, stream>>>(res_query, din_query,
      res_supp_fp, res_supp_bp, din_supp_fp, din_supp_bp, fg_thres, bg_thres, out, ws);
  k2_topk<<<dim3(4, B_), 256, 0, stream>>>(ws);
  k3_protos<<<dim3(C_, B_), 256, 0, stream>>>(res_query, din_query, ws);
  k4_gemm<<<dim3(32, 4), 32, 0, stream>>>(W_mean_fp, W_logvar_fp, W_mean_bp, W_logvar_bp,
      alpha_fp, beta_fp, alpha_bp, beta_bp, out, ws);
  k5_gaussian<<<dim3(2, B_), 512, 0, stream>>>(alpha_fp, beta_fp, alpha_bp, beta_bp, out, ws);
}